// SparseEmbedding_69011534512743
// MI455X (gfx1250) — compile-verified
//
#include <hip/hip_runtime.h>
#include <hip/hip_bf16.h>
#include <stdint.h>

// Embedding row gather: out[row, :] = weight[indices[row], :]
// C = 64 floats (256 B) per row in the reference; purely HBM-bound.
// Strategy: 4 threads per row, each thread moves 4 x float4 (b128) with
// interleaved segments for per-instruction coalescing; non-temporal stores
// for the streaming 52 MB output.

typedef float __attribute__((ext_vector_type(4))) v4f;

__global__ __launch_bounds__(256) void
sparse_embed_gather_c64(const long long* __restrict__ indices,
                        const float* __restrict__ weight,
                        float* __restrict__ out,
                        int n_rows) {
    int t   = blockIdx.x * 256 + threadIdx.x;
    int row = t >> 2;      // 4 threads per row
    int seg = t & 3;       // which float4 lane-group within the row
    if (row >= n_rows) return;

    long long idx = indices[row];

    const v4f* __restrict__ src = (const v4f*)(weight + (size_t)idx * 64);
    v4f*       __restrict__ dst = (v4f*)(out + (size_t)row * 64);

    // Interleaved segments: lanes (4k..4k+3) of a wave touch contiguous 64 B
    // chunks per instruction. 4 outstanding b128 loads per thread for MLP.
    v4f a = src[seg + 0];
    v4f b = src[seg + 4];
    v4f c = src[seg + 8];
    v4f d = src[seg + 12];

    // Output is write-once streaming data: non-temporal hint (TH=NT) so the
    // 52 MB result does not evict gathered weight rows from WGP$/L2.
    __builtin_nontemporal_store(a, dst + seg + 0);
    __builtin_nontemporal_store(b, dst + seg + 4);
    __builtin_nontemporal_store(c, dst + seg + 8);
    __builtin_nontemporal_store(d, dst + seg + 12);
}

// Fallback for arbitrary embedding_dim: one thread per output element.
__global__ __launch_bounds__(256) void
sparse_embed_gather_generic(const long long* __restrict__ indices,
                            const float* __restrict__ weight,
                            float* __restrict__ out,
                            long long total, int C) {
    long long i = (long long)blockIdx.x * blockDim.x + threadIdx.x;
    if (i >= total) return;
    long long row = i / C;
    int       col = (int)(i - row * (long long)C);
    long long idx = indices[row];
    out[i] = __builtin_nontemporal_load(weight + idx * (long long)C + col);
}

extern "C" void kernel_launch(void* const* d_in, const int* in_sizes, int n_in,
                              void* d_out, int out_size, void* d_ws, size_t ws_size,
                              hipStream_t stream) {
    // setup_inputs() order: indices (int64, B*H), weight (f32, NE*C),
    // exp_avgs (f32), exp_avg_sqs (f32). Optimizer state is unused in fwd.
    const long long* indices = (const long long*)d_in[0];
    const float*     weight  = (const float*)d_in[1];
    float*           out     = (float*)d_out;

    const int n_rows = in_sizes[0];                 // B*H = 204800
    const int C      = (n_rows > 0) ? (out_size / n_rows) : 0;

    if (C == 64) {
        // 4 threads/row -> 64 rows per 256-thread block (8 wave32 waves).
        const int rows_per_block = 64;
        const int grid = (n_rows + rows_per_block - 1) / rows_per_block;
        sparse_embed_gather_c64<<<grid, 256, 0, stream>>>(indices, weight, out, n_rows);
    } else if (C > 0) {
        const long long total = (long long)out_size;
        const int grid = (int)((total + 255) / 256);
        sparse_embed_gather_generic<<<grid, 256, 0, stream>>>(indices, weight, out, total, C);
    }
}